// DMPNNLastLayer_39118562132568
// MI455X (gfx1250) — compile-verified
//
#include <hip/hip_runtime.h>

#define N_NODES 40000
#define N_EDGES 640000
#define F_NODE  128
#define HIDDEN  128
#define KTOT    256   // F_NODE + HIDDEN
#define OUTF    128

typedef __attribute__((ext_vector_type(2))) float v2f;
typedef __attribute__((ext_vector_type(8))) float v8f;

// ---------------------------------------------------------------------------
// Device-scope hardware FP32 atomic add (avoids any CAS-loop fallback).
// ---------------------------------------------------------------------------
__device__ __forceinline__ void gatomic_add_f32(float* p, float v) {
    asm volatile("global_atomic_add_f32 %0, %1, off scope:SCOPE_DEV"
                 :: "v"(p), "v"(v)
                 : "memory");
}

// ---------------------------------------------------------------------------
// Zero the h_aggr accumulator (float4 per thread).
// ---------------------------------------------------------------------------
__global__ void __launch_bounds__(256)
zero_kernel(float4* __restrict__ p, int n4) {
    int i = blockIdx.x * 256 + threadIdx.x;
    if (i < n4) p[i] = make_float4(0.f, 0.f, 0.f, 0.f);
}

// ---------------------------------------------------------------------------
// Scatter-add: one wave per edge; each lane handles 4 contiguous floats.
// ---------------------------------------------------------------------------
__global__ void __launch_bounds__(256)
scatter_kernel(const float* __restrict__ h,
               const int*   __restrict__ edge_dst,
               float*       __restrict__ h_aggr) {
    int gid  = blockIdx.x * 256 + threadIdx.x;
    int e    = gid >> 5;          // edge index (one wave per edge)
    int lane = gid & 31;          // lane -> 4-float chunk
    if (e >= N_EDGES) return;

    int dst = edge_dst[e];
    float4 v = ((const float4*)(h + (size_t)e * HIDDEN))[lane];
    float* base = h_aggr + (size_t)dst * HIDDEN + lane * 4;
    gatomic_add_f32(base + 0, v.x);
    gatomic_add_f32(base + 1, v.y);
    gatomic_add_f32(base + 2, v.z);
    gatomic_add_f32(base + 3, v.w);
}

// ---------------------------------------------------------------------------
// Fused concat + GEMM (WMMA f32 16x16x4) + bias + ReLU.
// One wave handles 16 node rows x 128 output channels (8 C-tiles in VGPRs).
// q[n][k] = k < 128 ? x[n][k] : h_aggr[n][k-128];  out = relu(q @ W^T + b)
// ---------------------------------------------------------------------------
__global__ void __launch_bounds__(128)
gemm_kernel(const float* __restrict__ x,
            const float* __restrict__ h_aggr,
            const float* __restrict__ W,      // [128, 256] row-major
            const float* __restrict__ bias,   // [128]
            float*       __restrict__ out) {  // [40000, 128]
    const int lane  = threadIdx.x & 31;
    const int wib   = threadIdx.x >> 5;                 // wave in block (0..3)
    const int wave  = blockIdx.x * 4 + wib;             // 0..2499
    const int row0  = wave * 16;                        // 16 rows per wave
    const int half  = lane >> 4;                        // 0 or 1
    const int l16   = lane & 15;

    v8f acc[8];
    v8f zero = {0.f, 0.f, 0.f, 0.f, 0.f, 0.f, 0.f, 0.f};
#pragma unroll
    for (int t = 0; t < 8; ++t) acc[t] = zero;

    // A fragment row owned by this lane (ISA 16x4 f32 A layout):
    //   lanes 0-15 hold K = kb+0, kb+1 ; lanes 16-31 hold K = kb+2, kb+3
    const int arow = row0 + l16;
    const float* xrow = x      + (size_t)arow * F_NODE;
    const float* hrow = h_aggr + (size_t)arow * HIDDEN;

    // ---- K in [0,128): A from x ----
    for (int kb = 0; kb < F_NODE; kb += 4) {
        int ka = kb + 2 * half;
        v2f a = *(const v2f*)(xrow + ka);
#pragma unroll
        for (int t = 0; t < 8; ++t) {
            int col = t * 16 + l16;                      // output channel
            // B (4x16, K x N): lanes 0-15 rows kb+0/kb+1, lanes 16-31 kb+2/kb+3
            v2f b = *(const v2f*)(W + (size_t)col * KTOT + ka);
            acc[t] = __builtin_amdgcn_wmma_f32_16x16x4_f32(
                false, a, false, b, (short)0, acc[t], false, false);
        }
    }
    // ---- K in [128,256): A from h_aggr ----
    for (int kb = 0; kb < HIDDEN; kb += 4) {
        int ka = kb + 2 * half;
        v2f a = *(const v2f*)(hrow + ka);
#pragma unroll
        for (int t = 0; t < 8; ++t) {
            int col = t * 16 + l16;
            v2f b = *(const v2f*)(W + (size_t)col * KTOT + F_NODE + ka);
            acc[t] = __builtin_amdgcn_wmma_f32_16x16x4_f32(
                false, a, false, b, (short)0, acc[t], false, false);
        }
    }

    // ---- bias + ReLU + store (C/D layout: VGPR r -> M = r + 8*half, N = l16) ----
#pragma unroll
    for (int t = 0; t < 8; ++t) {
        int col = t * 16 + l16;
        float bv = bias[col];
#pragma unroll
        for (int r = 0; r < 8; ++r) {
            int row = row0 + half * 8 + r;
            float v = acc[t][r] + bv;
            v = v > 0.f ? v : 0.f;
            out[(size_t)row * OUTF + col] = v;
        }
    }
}

// ---------------------------------------------------------------------------
// Harness entry point.
// Inputs (setup_inputs order): x[40000,128] f32, h[640000,128] f32,
// edge_dst[640000] i32, W[128,256] f32, b[128] f32. Output: [40000,128] f32.
// ---------------------------------------------------------------------------
extern "C" void kernel_launch(void* const* d_in, const int* in_sizes, int n_in,
                              void* d_out, int out_size, void* d_ws, size_t ws_size,
                              hipStream_t stream) {
    const float* x        = (const float*)d_in[0];
    const float* h        = (const float*)d_in[1];
    const int*   edge_dst = (const int*)  d_in[2];
    const float* W        = (const float*)d_in[3];
    const float* bias     = (const float*)d_in[4];
    float*       out      = (float*)d_out;
    float*       h_aggr   = (float*)d_ws;   // 40000*128*4 = 20.48 MB scratch

    // 1) zero h_aggr
    const int n4 = (N_NODES * HIDDEN) / 4;          // 1,280,000 float4s
    zero_kernel<<<(n4 + 255) / 256, 256, 0, stream>>>((float4*)h_aggr, n4);

    // 2) scatter-add edges (1 wave per edge => 32 threads/edge)
    const long long sth = (long long)N_EDGES * 32;  // 20,480,000 threads
    scatter_kernel<<<(int)((sth + 255) / 256), 256, 0, stream>>>(h, edge_dst, h_aggr);

    // 3) fused concat + GEMM + bias + ReLU  (2500 waves, 4 waves/block)
    gemm_kernel<<<N_NODES / 16 / 4, 128, 0, stream>>>(x, h_aggr, W, bias, out);
}